// SimpleEdgeConvModel_83167746720200
// MI455X (gfx1250) — compile-verified
//
#include <hip/hip_runtime.h>
#include <hip/hip_bf16.h>

#define NPTS 16384
#define KNN  16
#define HDIM 64
#define CDIM 10
#define QW   8            // waves per kNN block / points per econv block

typedef __attribute__((ext_vector_type(2))) float v2f;
typedef __attribute__((ext_vector_type(8))) float v8f;

__device__ __forceinline__ v8f wmma4(v2f a, v2f b, v8f c) {
    // D = A(16x4 f32) * B(4x16 f32) + C(16x16 f32)
    return __builtin_amdgcn_wmma_f32_16x16x4_f32(
        /*neg_a=*/false, a, /*neg_b=*/false, b,
        /*c_mod=*/(short)0, c, /*reuse_a=*/false, /*reuse_b=*/false);
}

// Intra-wave LDS fence: DS ops from one wave are pipeline-ordered; we only
// need all stores complete + a compiler scheduling barrier before the
// cross-lane reads by the same wave.
__device__ __forceinline__ void lds_fence_wave() {
    asm volatile("s_wait_dscnt 0x0" ::: "memory");
    __builtin_amdgcn_wave_barrier();
}

// ---------------------------------------------------------------------------
// Kernel 1: kNN graph. 256-thread block = 8 waves, 16 queries per block.
// Each wave scans a disjoint 2048-candidate range; one V_WMMA_F32_16X16X4_F32
// Gram tile per 16 candidates. Selection uses an unsorted replace-worst
// top-16 (fast path: 1 LDS load + 1 compare, no register-array copies);
// all 32 lanes select (lane = query l16, candidates half*8..half*8+8).
// Ranking key |c|^2 - 2<q,c> (row-constant |q|^2 dropped; ordering unchanged).
// Exact (dist,index) ordering is restored in the merge phase.
// ---------------------------------------------------------------------------
__global__ __launch_bounds__(256) void knn_kernel(const float* __restrict__ pos,
                                                  int* __restrict__ knn_out) {
    const int qbase = blockIdx.x * 16;
    const int tid   = threadIdx.x;
    const int wave  = tid >> 5;
    const int lane  = tid & 31;
    const int half  = lane >> 4;
    const int l16   = lane & 15;

    __shared__ float tile[QW][16 * 16];        // per-wave Gram staging
    __shared__ float mdist[2 * QW * 16 * 16];  // [sublist 0..15][query][slot]
    __shared__ int   midx [2 * QW * 16 * 16];

    // A: 16x4 fp32 query tile (lanes 0-15: K=0,1; lanes 16-31: K=2,3; K3=0).
    const int qm = qbase + l16;
    const float q0 = pos[qm * 3 + 0];
    const float q1 = pos[qm * 3 + 1];
    const float q2 = pos[qm * 3 + 2];
    v2f a;
    a.x = (half == 0) ? q0 : q2;
    a.y = (half == 0) ? q1 : 0.0f;

    // Unsorted top-16 buffer + tracked worst (value, index, position).
    float best[16];
    int   bidx[16];
#pragma unroll
    for (int i = 0; i < 16; ++i) { best[i] = __builtin_inff(); bidx[i] = 0x7fffffff; }
    float wv = __builtin_inff();
    int   wi = 0x7fffffff;
    int   wp = 0;

    const int cstart = wave * (NPTS / QW);
    const int cend   = cstart + (NPTS / QW);
    for (int cb = cstart; cb < cend; cb += 16) {
        const int cm = cb + l16;
        const float c0 = pos[cm * 3 + 0];
        const float c1 = pos[cm * 3 + 1];
        const float c2 = pos[cm * 3 + 2];
        v2f b;
        b.x = (half == 0) ? c0 : c2;
        b.y = (half == 0) ? c1 : 0.0f;

        v8f acc = {};
        acc = wmma4(a, b, acc);                         // <q,c> tile

        const float sqc = c0 * c0 + c1 * c1 + c2 * c2;
#pragma unroll
        for (int r = 0; r < 8; ++r)
            tile[wave][(half * 8 + r) * 16 + l16] = sqc - 2.0f * acc[r];
        lds_fence_wave();

        // Selection: this lane owns query l16, candidates j = half*8 .. +8.
#pragma unroll 1
        for (int j = 0; j < 8; ++j) {
            const float d = tile[wave][l16 * 16 + half * 8 + j];
            if (d < wv) {                               // strict: keeps low idx on ties
                const int ci = cb + half * 8 + j;
                // Evict tracked worst slot (unrolled -> stays in VGPRs).
#pragma unroll
                for (int p = 0; p < 16; ++p)
                    if (p == wp) { best[p] = d; bidx[p] = ci; }
                // Rescan worst by lexicographic (value, index): evictions
                // always remove the highest-index among equal values.
                wv = best[0]; wi = bidx[0]; wp = 0;
#pragma unroll
                for (int p = 1; p < 16; ++p) {
                    const bool g = (best[p] > wv) ||
                                   (best[p] == wv && bidx[p] > wi);
                    if (g) { wv = best[p]; wi = bidx[p]; wp = p; }
                }
            }
        }
    }

    // Dump the 2 sublists per wave per query.
    const int sub = wave * 2 + half;                    // 0..15
#pragma unroll
    for (int s = 0; s < 16; ++s) {
        mdist[(sub * 16 + l16) * 16 + s] = best[s];
        midx [(sub * 16 + l16) * 16 + s] = bidx[s];
    }
    __syncthreads();

    // Merge: one thread per query; exact (dist, index) lexicographic order.
    if (tid < 16) {
        float fb[16]; int fi[16];
#pragma unroll
        for (int i = 0; i < 16; ++i) { fb[i] = __builtin_inff(); fi[i] = 0x7fffffff; }
#pragma unroll 1
        for (int w = 0; w < 2 * QW; ++w) {
#pragma unroll 1
            for (int s = 0; s < 16; ++s) {
                const float cd = mdist[(w * 16 + tid) * 16 + s];
                const int   ci = midx [(w * 16 + tid) * 16 + s];
                if (cd > fb[15] || (cd == fb[15] && ci >= fi[15])) continue;
                bool placed = false;
#pragma unroll
                for (int p = 15; p > 0; --p) {
                    if (!placed) {
                        const bool before = (cd < fb[p - 1]) ||
                                            (cd == fb[p - 1] && ci < fi[p - 1]);
                        if (before) { fb[p] = fb[p - 1]; fi[p] = fi[p - 1]; }
                        else        { fb[p] = cd; fi[p] = ci; placed = true; }
                    }
                }
                if (!placed) { fb[0] = cd; fi[0] = ci; }
            }
        }
#pragma unroll
        for (int s = 0; s < 16; ++s)
            knn_out[(qbase + tid) * KNN + s] = fi[s];
    }
}

// ---------------------------------------------------------------------------
// Kernel 2: edge_conv1 + ReLU. One wave per point (8 per block);
// M = 16 neighbors, K = 6 padded to 8 (2 WMMA steps), N = 64 (4 tiles).
// ---------------------------------------------------------------------------
__global__ __launch_bounds__(256) void econv1_kernel(const float* __restrict__ pos,
                                                     const int* __restrict__ knn_in,
                                                     const float* __restrict__ W1,
                                                     const float* __restrict__ b1,
                                                     float* __restrict__ h1) {
    const int i    = blockIdx.x * QW + (threadIdx.x >> 5);
    const int lane = threadIdx.x & 31;
    const int half = lane >> 4;
    const int l16  = lane & 15;

    const int j = knn_in[i * KNN + l16];          // this lane's A-row neighbor
    const float xi0 = pos[i * 3 + 0], xi1 = pos[i * 3 + 1], xi2 = pos[i * 3 + 2];
    const float d0 = pos[j * 3 + 0] - xi0;
    const float d1 = pos[j * 3 + 1] - xi1;
    const float d2 = pos[j * 3 + 2] - xi2;
    // msg row: [xi0,xi1,xi2, d0,d1,d2, 0, 0]
    v2f a0, a1;
    a0.x = (half == 0) ? xi0 : xi2;
    a0.y = (half == 0) ? xi1 : d0;
    a1.x = (half == 0) ? d1 : 0.0f;
    a1.y = (half == 0) ? d2 : 0.0f;

#pragma unroll
    for (int t = 0; t < 4; ++t) {
        const int n = t * 16 + l16;
        v2f b0, b1v;
        const int k0 = half * 2;                   // 0 or 2
        b0.x = W1[(k0 + 0) * HDIM + n];
        b0.y = W1[(k0 + 1) * HDIM + n];
        if (half == 0) {                           // W1 rows 4,5
            b1v.x = W1[4 * HDIM + n];
            b1v.y = W1[5 * HDIM + n];
        } else {                                   // padded rows 6,7
            b1v.x = 0.0f; b1v.y = 0.0f;
        }
        v8f acc = {};
        acc = wmma4(a0, b0, acc);
        acc = wmma4(a1, b1v, acc);

        float m = acc[0];
#pragma unroll
        for (int r = 1; r < 8; ++r) m = fmaxf(m, acc[r]);
        m = fmaxf(m, __shfl_xor(m, 16, 32));       // combine M halves
        if (lane < 16)
            h1[i * HDIM + n] = fmaxf(m + b1[n], 0.0f);
    }
}

// ---------------------------------------------------------------------------
// Kernel 2b: pack W2 so each econv2 lane reads its 64 weights per output tile
// as one contiguous 256 B run (b128-vectorizable).
// W2p[((t*32 + l)*32 + kc)*2 + e] = W2[(kc*4 + (l>>4)*2 + e)*64 + t*16 + (l&15)]
// ---------------------------------------------------------------------------
__global__ __launch_bounds__(256) void packW2_kernel(const float* __restrict__ W2,
                                                     float* __restrict__ W2p) {
    const int tid = blockIdx.x * 256 + threadIdx.x;   // 8192 total
    const int e  = tid & 1;
    const int kc = (tid >> 1) & 31;
    const int l  = (tid >> 6) & 31;
    const int t  = tid >> 11;
    const int k  = kc * 4 + (l >> 4) * 2 + e;
    const int n  = t * 16 + (l & 15);
    W2p[tid] = W2[k * HDIM + n];
}

// ---------------------------------------------------------------------------
// Kernel 3: edge_conv2 + ReLU. One wave per point (8 per block), fully
// register-blocked: A fragment (32 v2f) preloaded from h1, W2 fragment
// (32 v2f) preloaded from packed W2; inner loop is 32 register-only WMMAs
// per output tile split over two accumulators.
// ---------------------------------------------------------------------------
__global__ __launch_bounds__(256) void econv2_kernel(const float* __restrict__ h1,
                                                     const int* __restrict__ knn_in,
                                                     const float* __restrict__ W2p,
                                                     const float* __restrict__ b2,
                                                     float* __restrict__ h2) {
    const int i    = blockIdx.x * QW + (threadIdx.x >> 5);
    const int lane = threadIdx.x & 31;
    const int half = lane >> 4;
    const int l16  = lane & 15;

    const int j = knn_in[i * KNN + l16];          // A-row neighbor
    const float* hib = h1 + (size_t)i * HDIM;
    const float* hjb = h1 + (size_t)j * HDIM;

    // A fragments: K-chunks 0..15 carry h_i, 16..31 carry (h_j - h_i).
    v2f fhi[16], fdf[16];
#pragma unroll
    for (int kc = 0; kc < 16; ++kc) {
        const int c = kc * 4 + half * 2;
        const v2f va = *(const v2f*)(hib + c);
        const v2f vb = *(const v2f*)(hjb + c);
        fhi[kc] = va;
        fdf[kc].x = vb.x - va.x;
        fdf[kc].y = vb.y - va.y;
    }

#pragma unroll 1
    for (int t = 0; t < 4; ++t) {
        const int n = t * 16 + l16;
        const float* wb = W2p + ((size_t)t * 32 + lane) * 64;
        v2f wv[32];
#pragma unroll
        for (int kc = 0; kc < 32; ++kc)
            wv[kc] = *(const v2f*)(wb + kc * 2);

        v8f acc0 = {}, acc1 = {};
#pragma unroll
        for (int kc = 0; kc < 16; ++kc) acc0 = wmma4(fhi[kc], wv[kc], acc0);
#pragma unroll
        for (int kc = 0; kc < 16; ++kc) acc1 = wmma4(fdf[kc], wv[16 + kc], acc1);

        float m = acc0[0] + acc1[0];
#pragma unroll
        for (int r = 1; r < 8; ++r) m = fmaxf(m, acc0[r] + acc1[r]);
        m = fmaxf(m, __shfl_xor(m, 16, 32));
        if (lane < 16)
            h2[(size_t)i * HDIM + n] = fmaxf(m + b2[n], 0.0f);
    }
}

// ---------------------------------------------------------------------------
// Kernel 4: global channel max (ReLU => non-negative, init 0 exact) then
// out = g @ Wc + bc. Single block.
// ---------------------------------------------------------------------------
__global__ __launch_bounds__(1024) void head_kernel(const float* __restrict__ h2,
                                                    const float* __restrict__ Wc,
                                                    const float* __restrict__ bc,
                                                    float* __restrict__ out) {
    __shared__ float red[1024];
    const int t   = threadIdx.x;
    const int ch  = t & 63;
    const int grp = t >> 6;                        // 0..15

    float m = 0.0f;
    for (int r = grp; r < NPTS; r += 16) m = fmaxf(m, h2[(size_t)r * HDIM + ch]);
    red[t] = m;
    __syncthreads();

    if (t < HDIM) {
        float g = red[t];
#pragma unroll
        for (int k = 1; k < 16; ++k) g = fmaxf(g, red[t + k * 64]);
        red[t] = g;                                // g[ch]
    }
    __syncthreads();

    if (t < CDIM) {
        float s = bc[t];
#pragma unroll 1
        for (int h = 0; h < HDIM; ++h) s += red[h] * Wc[h * CDIM + t];
        out[t] = s;
    }
}

// ---------------------------------------------------------------------------
extern "C" void kernel_launch(void* const* d_in, const int* in_sizes, int n_in,
                              void* d_out, int out_size, void* d_ws, size_t ws_size,
                              hipStream_t stream) {
    const float* pos = (const float*)d_in[0];
    // d_in[1] = batch (all zeros) — single segment, unused
    const float* W1  = (const float*)d_in[2];
    const float* b1  = (const float*)d_in[3];
    const float* W2  = (const float*)d_in[4];
    const float* b2  = (const float*)d_in[5];
    const float* Wc  = (const float*)d_in[6];
    const float* bc  = (const float*)d_in[7];
    float* out = (float*)d_out;

    // Workspace: knn idx (1 MB) | h1 (4 MB) | h2 (4 MB) | packed W2 (32 KB)
    int*   d_knn = (int*)d_ws;
    float* d_h1  = (float*)((char*)d_ws + (size_t)NPTS * KNN * sizeof(int));
    float* d_h2  = d_h1 + (size_t)NPTS * HDIM;
    float* d_W2p = d_h2 + (size_t)NPTS * HDIM;

    packW2_kernel<<<32,          256, 0, stream>>>(W2, d_W2p);
    knn_kernel   <<<NPTS / 16,   256, 0, stream>>>(pos, d_knn);
    econv1_kernel<<<NPTS / QW,   256, 0, stream>>>(pos, d_knn, W1, b1, d_h1);
    econv2_kernel<<<NPTS / QW,   256, 0, stream>>>(d_h1, d_knn, d_W2p, b2, d_h2);
    head_kernel  <<<1,          1024, 0, stream>>>(d_h2, Wc, bc, out);
}